// UDTCWTDirect_32152125178187
// MI455X (gfx1250) — compile-verified
//
#include <hip/hip_runtime.h>

// Fused 8-level undecimated dual-tree CWT, one pass over HBM.
// Layout: d_out = [ yl (4*8*65536) | yh (8*4*16*65536) ], float32.

#define T_LEN   65536
#define NB      4
#define NC      8
#define TILE    2048
#define NTHR    256
// halo: levels 2..8 need sum_{j=1..7} 9*2^j/2 = 1143 each side; level-1 taps add <=3.
#define X0LEN   (TILE + 2292)   // raw x tile + halo (base position t0-1146)
#define L1LEN   (TILE + 2286)   // level-1 phi output (base position t0-1143)
#define YL_SZ   ((size_t)NB * NC * T_LEN)

typedef float f32x2 __attribute__((ext_vector_type(2), aligned(8)));
typedef f32x2 __attribute__((may_alias)) f32x2_a;

#if defined(__HIP_DEVICE_COMPILE__) && defined(__gfx1250__) && \
    __has_builtin(__builtin_amdgcn_global_load_async_to_lds_b32)
#define USE_ASYNC_LDS 1
typedef __attribute__((address_space(1))) int as1_int;  // global
typedef __attribute__((address_space(3))) int as3_int;  // LDS
__device__ __forceinline__ void wait_asynccnt0() {
#if __has_builtin(__builtin_amdgcn_s_wait_asynccnt)
  __builtin_amdgcn_s_wait_asynccnt(0);
#else
  asm volatile("s_wait_asynccnt 0" ::: "memory");
#endif
}
#else
#define USE_ASYNC_LDS 0
#endif

__global__ __launch_bounds__(NTHR) void udtcwt_fused_kernel(
    const float* __restrict__ x,
    const float* __restrict__ h0o, const float* __restrict__ h1o,
    const float* __restrict__ h0a, const float* __restrict__ h1a,
    const float* __restrict__ h0b, const float* __restrict__ h1b,
    float* __restrict__ out)
{
  __shared__ __attribute__((aligned(16))) float sx [X0LEN];
  __shared__ __attribute__((aligned(16))) float sa0[L1LEN];  // derived channel c
  __shared__ __attribute__((aligned(16))) float sa1[L1LEN];  // derived channel c+8

  const int tid = threadIdx.x;
  const int t0  = blockIdx.x * TILE;
  const int bc  = blockIdx.y;          // b*8 + c
  const int b   = bc >> 3;
  const int c   = bc & 7;

  const float* __restrict__ xrow = x + (size_t)bc * T_LEN;

  // ---- filter coefficients (uniform -> SGPR loads) ----
  float c0[5], c1[7], fa[10], fb[10];
#pragma unroll
  for (int k = 0; k < 5; ++k) c0[k] = h0o[k];
#pragma unroll
  for (int k = 0; k < 7; ++k) c1[k] = h1o[k];
  {
    // even derived channels -> h?a, odd -> h?b; channels c and c+8 share parity.
    const float* __restrict__ pa = (c & 1) ? h0b : h0a;
    const float* __restrict__ pb = (c & 1) ? h1b : h1a;
#pragma unroll
    for (int k = 0; k < 10; ++k) { fa[k] = pa[k]; fb[k] = pb[k]; }
  }

  // ---- stage x tile + halo into LDS (zero outside [0,T)) ----
  // Interior lanes use the CDNA5 async global->LDS copy path (ASYNCcnt);
  // out-of-range lanes zero-fill their LDS slot directly.
  for (int u = tid; u < X0LEN; u += NTHR) {
    const int g = t0 - 1146 + u;
    const bool ok = (g >= 0) && (g < T_LEN);
#if USE_ASYNC_LDS
    if (ok) {
      __builtin_amdgcn_global_load_async_to_lds_b32(
          (as1_int*)(xrow + g), (as3_int*)(sx + u), 0, 0);
    } else {
      sx[u] = 0.0f;
    }
#else
    sx[u] = ok ? xrow[g] : 0.0f;
#endif
  }
#if USE_ASYNC_LDS
  wait_asynccnt0();
#endif
  __syncthreads();

  // ---- level 1: x -> phi (kept in LDS) and psi (written out, channel-swapped) ----
  // position of index i: t = t0 - 1143 + i ; sx base = t0 - 1146
  for (int i = tid; i < L1LEN; i += NTHR) {
    float s0 = 0.0f, s1 = 0.0f;          // phi unshifted / shifted(+1)
#pragma unroll
    for (int k = 0; k < 5; ++k) {
      const float w = c0[k];
      s0 = fmaf(sx[i + 1 + k], w, s0);   // x[t-2+k]
      s1 = fmaf(sx[i + 2 + k], w, s1);   // x[t-1+k]
    }
    sa0[i] = s0;
    sa1[i] = s1;

    const int t = i - 1143;
    if ((unsigned)t < (unsigned)TILE) {  // psi only on the central tile
      float p0 = 0.0f, p1 = 0.0f;
#pragma unroll
      for (int k = 0; k < 7; ++k) {
        const float w = c1[k];
        p0 = fmaf(sx[i + 1 + k], w, p0); // shifted branch -> out channel c
        p1 = fmaf(sx[i + k],     w, p1); // unshifted      -> out channel c+8
      }
      const size_t o0 = YL_SZ + ((size_t)(b * 16 + c)) * T_LEN + (size_t)(t0 + t);
      __builtin_nontemporal_store(p0, out + o0);
      __builtin_nontemporal_store(p1, out + o0 + (size_t)8 * T_LEN);
    }
  }
  __syncthreads();

  // ---- levels 2..8: in-place dilated depthwise conv chain in LDS ----
  // invariant: out[i] = sum_k in[i + k*d] * h[k]  (array base shifts by 9*d/2 per level)
  for (int j = 1; j <= 7; ++j) {
    const int d    = 1 << j;
    const int Hn   = 9 * (128 - d);          // halo remaining AFTER this level (even)
    const int Lout = TILE + 2 * Hn;          // even
    const size_t yb0 = YL_SZ + ((size_t)((j * NB + b) * 16 + c)) * T_LEN + (size_t)t0;
    const size_t yb1 = yb0 + (size_t)8 * T_LEN;

    for (int base = 0; base < Lout; base += 2 * NTHR) {
      const int  i0  = base + 2 * tid;       // even -> aligned b64 LDS accesses
      const bool act = (i0 < Lout);
      float r00 = 0.f, r01 = 0.f, r10 = 0.f, r11 = 0.f;   // phi pair, both channels
      if (act) {
        float p00 = 0.f, p01 = 0.f, p10 = 0.f, p11 = 0.f; // psi pair, both channels
#pragma unroll
        for (int k = 0; k < 10; ++k) {
          const int u = i0 + k * d;          // even
          const f32x2 va = *(const f32x2_a*)(sa0 + u);
          const f32x2 vb = *(const f32x2_a*)(sa1 + u);
          const float wa = fa[k], wb = fb[k];
          r00 = fmaf(va.x, wa, r00); r01 = fmaf(va.y, wa, r01);
          r10 = fmaf(vb.x, wa, r10); r11 = fmaf(vb.y, wa, r11);
          p00 = fmaf(va.x, wb, p00); p01 = fmaf(va.y, wb, p01);
          p10 = fmaf(vb.x, wb, p10); p11 = fmaf(vb.y, wb, p11);
        }
        const int t = i0 - Hn;               // even; pair fully in or out of tile
        if ((unsigned)t < (unsigned)TILE) {
          f32x2 q0; q0.x = p00; q0.y = p01;
          f32x2 q1; q1.x = p10; q1.y = p11;
          __builtin_nontemporal_store(q0, (f32x2_a*)(out + yb0 + t));
          __builtin_nontemporal_store(q1, (f32x2_a*)(out + yb1 + t));
        }
      }
      __syncthreads();                       // all reads of this chunk done
      if (act) {
        f32x2 w0; w0.x = r00; w0.y = r01;
        f32x2 w1; w1.x = r10; w1.y = r11;
        *(f32x2_a*)(sa0 + i0) = w0;
        *(f32x2_a*)(sa1 + i0) = w1;
      }
      __syncthreads();                       // writes visible before next chunk reads
    }
  }

  // ---- yl: final phi, first 8 channels only (derived channel c) ----
  for (int i = tid; i < TILE; i += NTHR) {
    __builtin_nontemporal_store(sa0[i], out + (size_t)bc * T_LEN + (size_t)(t0 + i));
  }
}

extern "C" void kernel_launch(void* const* d_in, const int* in_sizes, int n_in,
                              void* d_out, int out_size, void* d_ws, size_t ws_size,
                              hipStream_t stream) {
  (void)in_sizes; (void)n_in; (void)out_size; (void)d_ws; (void)ws_size;
  const float* x   = (const float*)d_in[0];
  const float* h0o = (const float*)d_in[1];
  const float* h1o = (const float*)d_in[2];
  const float* h0a = (const float*)d_in[3];
  const float* h1a = (const float*)d_in[4];
  const float* h0b = (const float*)d_in[5];
  const float* h1b = (const float*)d_in[6];
  float* out = (float*)d_out;

  dim3 grid(T_LEN / TILE, NB * NC);   // 32 tiles x 32 (b,c) pairs
  udtcwt_fused_kernel<<<grid, NTHR, 0, stream>>>(x, h0o, h1o, h0a, h1a, h0b, h1b, out);
}